// IPTBboxEmbedding_42880953484128
// MI455X (gfx1250) — compile-verified
//
#include <hip/hip_runtime.h>
#include <stdint.h>

typedef __attribute__((ext_vector_type(4))) float v4f;

#define IPT_VOCAB 50257
#define BBOX_NUM  1001
#define OCR_NUM   2
#define MAX_POS   4096
#define D         1024
#define D4        (D / 4)       // 256 float4 per embedding row
#define THREADS   256           // one float4 per lane covers a full row
#define TOK_PER_BLK 8

// CDNA5 async copy: global memory -> LDS, 16 bytes per lane, tracked by ASYNCcnt.
__device__ __forceinline__ void async_copy_b128(uint32_t lds_off, const void* gaddr) {
    asm volatile("global_load_async_to_lds_b128 %0, %1, off"
                 :: "v"(lds_off), "v"(gaddr)
                 : "memory");
}
__device__ __forceinline__ void wait_async_le2() {
    asm volatile("s_wait_asynccnt 0x2" ::: "memory");
}
__device__ __forceinline__ void wait_async_le0() {
    asm volatile("s_wait_asynccnt 0x0" ::: "memory");
}

__global__ void __launch_bounds__(THREADS)
ipt_bbox_embed_kernel(const int* __restrict__ tokens,
                      const int* __restrict__ pos_ids,
                      const float* __restrict__ word_emb,
                      const float* __restrict__ pos_emb,
                      const float* __restrict__ bbox_emb,
                      const float* __restrict__ ocr_emb,
                      float* __restrict__ out,
                      int n_tokens)
{
    // Double-buffered staging: [stage][rowA/rowB][lane] -> 2*2*256*16B = 16 KB LDS
    __shared__ v4f sbuf[2][2][D4];

    const int lane = threadIdx.x;               // 0..255, owns 16B of each row
    const int base = blockIdx.x * TOK_PER_BLK;

    // Raw 32-bit LDS byte offsets for the async engine (low 32 bits of the
    // generic address are the wave-relative LDS offset on gfx1250).
    uint32_t lds_a[2], lds_b[2];
    lds_a[0] = (uint32_t)(uintptr_t)&sbuf[0][0][lane];
    lds_b[0] = (uint32_t)(uintptr_t)&sbuf[0][1][lane];
    lds_a[1] = (uint32_t)(uintptr_t)&sbuf[1][0][lane];
    lds_b[1] = (uint32_t)(uintptr_t)&sbuf[1][1][lane];

    // Resolve token -> (rowA, rowB, flag) with out = rowA + flag * rowB.
    auto resolve = [&](int t, const float*& a, const float*& b, float& flag) {
        if (t >= n_tokens) { a = word_emb; b = word_emb; flag = 0.0f; return; }
        int tok = tokens[t];
        if (tok < 0) tok = 0;
        if (tok < IPT_VOCAB) {
            int p = pos_ids[t];
            if (p < 0) p = 0;
            if (p >= MAX_POS) p = MAX_POS - 1;
            a = word_emb + (size_t)tok * D;
            b = pos_emb + (size_t)p * D;
            flag = 1.0f;
        } else if (tok < IPT_VOCAB + BBOX_NUM) {
            const float* r = bbox_emb + (size_t)(tok - IPT_VOCAB) * D;
            a = r; b = r;           // duplicate slot: L2/L0 hit, result = a + 0*b
            flag = 0.0f;
        } else {
            int o = tok - IPT_VOCAB - BBOX_NUM;
            if (o >= OCR_NUM) o = OCR_NUM - 1;
            const float* r = ocr_emb + (size_t)o * D;
            a = r; b = r;
            flag = 0.0f;
        }
    };

    // Issue both async row copies for token t into LDS stage `st`; return flag.
    auto issue = [&](int t, int st) -> float {
        const float* a; const float* b; float flag;
        resolve(t, a, b, flag);
        async_copy_b128(lds_a[st], (const void*)(a + lane * 4));
        async_copy_b128(lds_b[st], (const void*)(b + lane * 4));
        return flag;
    };

    float flags[2];
    flags[0] = issue(base + 0, 0);

    #pragma unroll
    for (int i = 0; i < TOK_PER_BLK; ++i) {
        const int st = i & 1;
        if (i + 1 < TOK_PER_BLK) {
            flags[(i + 1) & 1] = issue(base + i + 1, (i + 1) & 1);
            wait_async_le2();   // in-order completion: <=2 in flight == stage st done
        } else {
            wait_async_le0();
        }
        const int t = base + i;
        if (t < n_tokens) {
            v4f va = sbuf[st][0][lane];     // ds_load_b128 (own 16B, no barrier)
            v4f vb = sbuf[st][1][lane];
            v4f r  = va + flags[st] * vb;
            // Streamed 128 MiB output: non-temporal so the 192 MB L2 keeps the
            // pos/bbox/ocr tables resident instead.
            __builtin_nontemporal_store(r, (v4f*)(out + (size_t)t * D) + lane);
        }
    }
}

extern "C" void kernel_launch(void* const* d_in, const int* in_sizes, int n_in,
                              void* d_out, int out_size, void* d_ws, size_t ws_size,
                              hipStream_t stream) {
    const int*   tokens   = (const int*)d_in[0];
    const int*   pos_ids  = (const int*)d_in[1];
    const float* word_emb = (const float*)d_in[2];
    const float* pos_emb  = (const float*)d_in[3];
    const float* bbox_emb = (const float*)d_in[4];
    const float* ocr_emb  = (const float*)d_in[5];
    float*       out      = (float*)d_out;

    const int n_tokens = in_sizes[0];   // B*S = 32768
    const int blocks   = (n_tokens + TOK_PER_BLK - 1) / TOK_PER_BLK;

    ipt_bbox_embed_kernel<<<blocks, THREADS, 0, stream>>>(
        tokens, pos_ids, word_emb, pos_emb, bbox_emb, ocr_emb, out, n_tokens);
}